// GridEncoder_24988119728172
// MI455X (gfx1250) — compile-verified
//
#include <hip/hip_runtime.h>
#include <stdint.h>

// ---------------------------------------------------------------------------
// GridEncoder forward (binarized STE hash grid), MI455X / gfx1250.
//
//  1) pack_level_kernel<ROW_OFF,HMAP> x12: params (4,078,528 x 2 f32, 32.6MB)
//     -> 2-bit sign table (~1MB) in d_ws. All level constants are immediates.
//  2) encode_dense_kernel: the 5 dense levels' packed tables (contiguous
//     102KB at ws words [0,25536)) are async-copied to LDS in one pass
//     (global_load_async_to_lds_b128 / ASYNCcnt / s_wait_asynccnt); each
//     point then does 40 LDS corner gathers and stores out cols 0..9 as
//     float4+float4+float2.
//  3) encode_hashed_kernel<RES> x7: one 128KB hashed table per kernel staged
//     to LDS the same way; hmap == 2^19 so the modulo is a mask.
// Output layout: out[p*24 + 2*l + f]  (level-major, matches reference).
// ---------------------------------------------------------------------------

static constexpr int NLEVELS          = 12;
static constexpr int DENSE_WORDS      = 25536;    // words 0..25536 = levels 0..4
static constexpr int HASH_WORDS       = 32768;    // 128KB per hashed level
static constexpr uint32_t HASH_MASK   = 524287u;  // 2^19 - 1
static constexpr uint32_t PRIME_Y     = 2654435761u;
static constexpr uint32_t PRIME_Z     = 805459861u;

// Packed-word base of each level in d_ws (padded to 16B boundaries).
static constexpr int WB[NLEVELS] = {0, 256, 1020, 3068, 9152, 25536, 58304,
                                    91072, 123840, 156608, 189376, 222144};

// --------------------------- pack kernels ----------------------------------
// One thread per packed word: 16 table rows -> 32 sign bits.
// bit(2j) = sign(f0) >= 0 ; bit(2j+1) = sign(f1) >= 0  (ste_binary: p>=0 -> +1)
template <int ROW_OFF, int HMAP>
__global__ __launch_bounds__(256) void pack_level_kernel(
    const float* __restrict__ params, uint32_t* __restrict__ packedLvl)
{
  constexpr int WORDS = (HMAP + 15) / 16;
  const int w = blockIdx.x * 256 + (int)threadIdx.x;
  if (w >= WORDS) return;

  const float2* rows = (const float2*)params + ROW_OFF;  // 2 feats/row, 8B aligned
  const int e0 = w * 16;
  uint32_t bits = 0u;
  #pragma unroll
  for (int j = 0; j < 16; ++j) {
    const int e = e0 + j;
    if (e < HMAP) {                       // immediate compare; tail word only
      const float2 f = rows[e];
      bits |= (f.x >= 0.f ? 1u : 0u) << (2 * j);
      bits |= (f.y >= 0.f ? 1u : 0u) << (2 * j + 1);
    }
  }
  packedLvl[w] = bits;
}

// ------------------------ async LDS staging --------------------------------
__device__ __forceinline__ void stage_to_lds(uint32_t* lut, const uint32_t* src,
                                             uint32_t nbytes)
{
  const uint64_t gbase = (uint64_t)(uintptr_t)src;        // SGPR pair
  const uint32_t lbase = (uint32_t)(uintptr_t)lut;        // LDS offset (low 32b)
  for (uint32_t goff = threadIdx.x * 16u; goff < nbytes; goff += 256u * 16u) {
    const uint32_t loff = lbase + goff;
    // GVS mode: mem = SGPR64 + VGPR32 ; LDS dest byte address in VDST VGPR.
    asm volatile("global_load_async_to_lds_b128 %0, %1, %2"
                 :: "v"(loff), "v"(goff), "s"(gbase) : "memory");
  }
  asm volatile("s_wait_asynccnt 0x0" ::: "memory");       // own wave's copies
  __syncthreads();                                        // LDS visible WG-wide
}

// ------------------------ per-level accumulators ---------------------------
template <int RES, int WBASE>
__device__ __forceinline__ void accum_dense(const uint32_t* lut,
                                            float px, float py, float pz,
                                            float* acc)
{
  const float sx = px * (float)(RES - 1);
  const float sy = py * (float)(RES - 1);
  const float sz = pz * (float)(RES - 1);
  const float flx = floorf(sx), fly = floorf(sy), flz = floorf(sz);
  const int ix = (int)flx, iy = (int)fly, iz = (int)flz;
  const float fx = sx - flx, fy = sy - fly, fz = sz - flz;

  const int cx0 = ix < RES - 1 ? ix : RES - 1, cx1 = ix + 1 < RES - 1 ? ix + 1 : RES - 1;
  const int cy0 = iy < RES - 1 ? iy : RES - 1, cy1 = iy + 1 < RES - 1 ? iy + 1 : RES - 1;
  const int cz0 = iz < RES - 1 ? iz : RES - 1, cz1 = iz + 1 < RES - 1 ? iz + 1 : RES - 1;

  const uint32_t dx[2] = {(uint32_t)cx0, (uint32_t)cx1};
  const uint32_t dy[2] = {(uint32_t)(cy0 * RES), (uint32_t)(cy1 * RES)};
  const uint32_t dz[2] = {(uint32_t)(cz0 * RES * RES), (uint32_t)(cz1 * RES * RES)};
  const float wx[2] = {1.f - fx, fx}, wy[2] = {1.f - fy, fy}, wz[2] = {1.f - fz, fz};

  float a0 = 0.f, a1 = 0.f;
  #pragma unroll
  for (int c = 0; c < 8; ++c) {
    const int bx = c & 1, by = (c >> 1) & 1, bz = (c >> 2) & 1;
    const uint32_t idx = dx[bx] + dy[by] + dz[bz];
    const float w = wx[bx] * wy[by] * wz[bz];
    const uint32_t b = lut[WBASE + (idx >> 4)] >> ((idx & 15u) << 1);
    a0 += (b & 1u) ? w : -w;
    a1 += (b & 2u) ? w : -w;
  }
  acc[0] = a0; acc[1] = a1;
}

template <int RES>
__device__ __forceinline__ void accum_hashed(const uint32_t* lut,
                                             float px, float py, float pz,
                                             float* acc)
{
  const float sx = px * (float)(RES - 1);
  const float sy = py * (float)(RES - 1);
  const float sz = pz * (float)(RES - 1);
  const float flx = floorf(sx), fly = floorf(sy), flz = floorf(sz);
  const int ix = (int)flx, iy = (int)fly, iz = (int)flz;
  const float fx = sx - flx, fy = sy - fly, fz = sz - flz;

  const int cx0 = ix < RES - 1 ? ix : RES - 1, cx1 = ix + 1 < RES - 1 ? ix + 1 : RES - 1;
  const int cy0 = iy < RES - 1 ? iy : RES - 1, cy1 = iy + 1 < RES - 1 ? iy + 1 : RES - 1;
  const int cz0 = iz < RES - 1 ? iz : RES - 1, cz1 = iz + 1 < RES - 1 ? iz + 1 : RES - 1;

  const uint32_t hx[2] = {(uint32_t)cx0, (uint32_t)cx1};                 // prime 1
  const uint32_t hy[2] = {(uint32_t)cy0 * PRIME_Y, (uint32_t)cy1 * PRIME_Y};
  const uint32_t hz[2] = {(uint32_t)cz0 * PRIME_Z, (uint32_t)cz1 * PRIME_Z};
  const float wx[2] = {1.f - fx, fx}, wy[2] = {1.f - fy, fy}, wz[2] = {1.f - fz, fz};

  float a0 = 0.f, a1 = 0.f;
  #pragma unroll
  for (int c = 0; c < 8; ++c) {
    const int bx = c & 1, by = (c >> 1) & 1, bz = (c >> 2) & 1;
    const uint32_t idx = (hx[bx] ^ hy[by] ^ hz[bz]) & HASH_MASK;
    const float w = wx[bx] * wy[by] * wz[bz];
    const uint32_t b = lut[idx >> 4] >> ((idx & 15u) << 1);
    a0 += (b & 1u) ? w : -w;
    a1 += (b & 2u) ? w : -w;
  }
  acc[0] = a0; acc[1] = a1;
}

// --------------------------- encode kernels --------------------------------
// Fused dense levels 0..4: one 102KB async LDS stage, 40 gathers/point,
// contiguous 40B store per point (cols 0..9).
__global__ __launch_bounds__(256) void encode_dense_kernel(
    const float* __restrict__ x, const uint32_t* __restrict__ packed,
    float* __restrict__ out, const int nPoints, const int ppb)
{
  __shared__ uint32_t lut[DENSE_WORDS];
  stage_to_lds(lut, packed, DENSE_WORDS * 4u);

  const int pStart = blockIdx.x * ppb + (int)threadIdx.x;
  int pEnd = blockIdx.x * ppb + ppb;
  if (pEnd > nPoints) pEnd = nPoints;

  for (int p = pStart; p < pEnd; p += 256) {
    const float px = x[3 * p + 0], py = x[3 * p + 1], pz = x[3 * p + 2];
    float acc[10];
    accum_dense<16, 0>   (lut, px, py, pz, acc + 0);
    accum_dense<23, 256> (lut, px, py, pz, acc + 2);
    accum_dense<32, 1020>(lut, px, py, pz, acc + 4);
    accum_dense<46, 3068>(lut, px, py, pz, acc + 6);
    accum_dense<64, 9152>(lut, px, py, pz, acc + 8);
    float* row = out + p * 24;                        // 16B-aligned (96B rows)
    *(float4*)(row + 0) = make_float4(acc[0], acc[1], acc[2], acc[3]);
    *(float4*)(row + 4) = make_float4(acc[4], acc[5], acc[6], acc[7]);
    *(float2*)(row + 8) = make_float2(acc[8], acc[9]);
  }
}

template <int RES>
__global__ __launch_bounds__(256) void encode_hashed_kernel(
    const float* __restrict__ x, const uint32_t* __restrict__ packedLvl,
    float* __restrict__ out, const int outCol, const int nPoints, const int ppb)
{
  __shared__ uint32_t lut[HASH_WORDS];                // 128KB
  stage_to_lds(lut, packedLvl, HASH_WORDS * 4u);

  const int pStart = blockIdx.x * ppb + (int)threadIdx.x;
  int pEnd = blockIdx.x * ppb + ppb;
  if (pEnd > nPoints) pEnd = nPoints;

  for (int p = pStart; p < pEnd; p += 256) {
    const float px = x[3 * p + 0], py = x[3 * p + 1], pz = x[3 * p + 2];
    float acc[2];
    accum_hashed<RES>(lut, px, py, pz, acc);
    *(float2*)(out + p * 24 + outCol) = make_float2(acc[0], acc[1]);  // 8B aligned
  }
}

// --------------------------- launcher --------------------------------------
extern "C" void kernel_launch(void* const* d_in, const int* in_sizes, int n_in,
                              void* d_out, int out_size, void* d_ws, size_t ws_size,
                              hipStream_t stream)
{
  (void)n_in; (void)out_size; (void)ws_size;
  const float* x      = (const float*)d_in[0];
  const float* params = (const float*)d_in[1];
  float*       out    = (float*)d_out;
  uint32_t*    packed = (uint32_t*)d_ws;       // needs ~1.02 MB of scratch
  const int nPoints   = in_sizes[0] / 3;

  // ---- pack: 12 tiny launches, all level constants are immediates ----
#define PACK(OFF, HMAP, L)                                                    \
  pack_level_kernel<OFF, HMAP>                                                \
      <<<(((HMAP + 15) / 16) + 255) / 256, 256, 0, stream>>>(params, packed + WB[L])
  PACK(0,       4096,   0);
  PACK(4096,    12168,  1);
  PACK(16264,   32768,  2);
  PACK(49032,   97336,  3);
  PACK(146368,  262144, 4);
  PACK(408512,  524288, 5);
  PACK(932800,  524288, 6);
  PACK(1457088, 524288, 7);
  PACK(1981376, 524288, 8);
  PACK(2505664, 524288, 9);
  PACK(3029952, 524288, 10);
  PACK(3554240, 524288, 11);
#undef PACK

  const int ppb  = 2048;                        // points per block
  const int grid = (nPoints + ppb - 1) / ppb;

  encode_dense_kernel<<<grid, 256, 0, stream>>>(x, packed, out, nPoints, ppb);

  encode_hashed_kernel<92> <<<grid, 256, 0, stream>>>(x, packed + WB[5],  out, 10, nPoints, ppb);
  encode_hashed_kernel<128><<<grid, 256, 0, stream>>>(x, packed + WB[6],  out, 12, nPoints, ppb);
  encode_hashed_kernel<184><<<grid, 256, 0, stream>>>(x, packed + WB[7],  out, 14, nPoints, ppb);
  encode_hashed_kernel<256><<<grid, 256, 0, stream>>>(x, packed + WB[8],  out, 16, nPoints, ppb);
  encode_hashed_kernel<368><<<grid, 256, 0, stream>>>(x, packed + WB[9],  out, 18, nPoints, ppb);
  encode_hashed_kernel<512><<<grid, 256, 0, stream>>>(x, packed + WB[10], out, 20, nPoints, ppb);
  encode_hashed_kernel<736><<<grid, 256, 0, stream>>>(x, packed + WB[11], out, 22, nPoints, ppb);
}